// FrequencyAnalyzer_60301340836470
// MI455X (gfx1250) — compile-verified
//
#include <hip/hip_runtime.h>
#include <hip/hip_bf16.h>
#include <math.h>

typedef __attribute__((ext_vector_type(2))) float v2f;
typedef __attribute__((ext_vector_type(8))) float v8f;

#define NPIX 224
#define PI_F 3.14159265358979323846f

// ---------------------------------------------------------------------------
// Kernel 1: per-frame grayscale + 4-row DCT projection + 128-coef extraction.
// grid = 256 frames, block = 224 threads (thread = image column w).
// Streams the 154 MB input once (coalesced across w), so it runs at HBM speed.
// ---------------------------------------------------------------------------
__global__ void __launch_bounds__(NPIX)
k_gray_dct_feats(const float* __restrict__ img, float* __restrict__ feats) {
    __shared__ float Ms[4][NPIX];   // DCT rows i = 0, 1, 32, 33
    __shared__ float Ps[4][NPIX];   // P = M_sel @ gray  (4 x 224)

    const int bt  = blockIdx.x;     // frame index (B*T = 256)
    const int tid = threadIdx.x;    // 0..223, owns column w = tid
    const float s = sqrtf(2.0f / (float)NPIX);

    // Precompute the 4 needed DCT basis rows (i=0 row carries the 1/sqrt2).
    {
        const int h = tid;
        const float ang = PI_F * (2.0f * (float)h + 1.0f) / (2.0f * (float)NPIX);
        Ms[0][h] = s * 0.70710678118654752f;     // i = 0: cos(0)=1, ortho scale
        Ms[1][h] = s * cosf(ang * 1.0f);
        Ms[2][h] = s * cosf(ang * 32.0f);
        Ms[3][h] = s * cosf(ang * 33.0f);
    }
    __syncthreads();

    // Stream the frame row by row; thread w accumulates P[0..3][w] in registers.
    const int w = tid;
    const float* __restrict__ base = img + (size_t)bt * 3u * NPIX * NPIX;
    float p0 = 0.f, p1 = 0.f, p2 = 0.f, p3 = 0.f;
    #pragma unroll 4
    for (int h = 0; h < NPIX; ++h) {
        const int off = h * NPIX + w;
        float r = base[off];
        float g = base[NPIX * NPIX + off];
        float b = base[2 * NPIX * NPIX + off];
        // (x*255).astype(uint8): truncation (inputs in [0,1) -> floor of clip)
        float ur = floorf(fminf(fmaxf(r * 255.0f, 0.0f), 255.0f));
        float ug = floorf(fminf(fmaxf(g * 255.0f, 0.0f), 255.0f));
        float ub = floorf(fminf(fmaxf(b * 255.0f, 0.0f), 255.0f));
        // cv2 RGB2GRAY weights, rounded (RNE matches jnp.round)
        float gray = rintf(fmaf(0.299f, ur, fmaf(0.587f, ug, 0.114f * ub)));
        p0 = fmaf(gray, Ms[0][h], p0);
        p1 = fmaf(gray, Ms[1][h], p1);
        p2 = fmaf(gray, Ms[2][h], p2);
        p3 = fmaf(gray, Ms[3][h], p3);
    }
    Ps[0][w] = p0; Ps[1][w] = p1; Ps[2][w] = p2; Ps[3][w] = p3;
    __syncthreads();

    // 128 features: f -> (row-select, DCT column j), then feat = P[row]·M[j,:]
    if (tid < 128) {
        const int f = tid;
        int isel, j;
        if (f < 112) {                       // block_i = 0: rows 0,1
            isel = (f >> 3) & 1;             // r
            j    = (f >> 4) * 32 + (f & 7);  // 32*bj + c
        } else {                             // block_i = 1, bj = 0: rows 32,33
            isel = 2 + (((f - 112) >> 3) & 1);
            j    = f & 7;
        }
        const float sj = s * (j == 0 ? 0.70710678118654752f : 1.0f);
        float acc = 0.f;
        for (int ww = 0; ww < NPIX; ++ww) {
            float ang = PI_F * (2.0f * (float)ww + 1.0f) * (float)j
                        / (2.0f * (float)NPIX);
            acc = fmaf(Ps[isel][ww], sj * cosf(ang), acc);
        }
        feats[bt * 128 + f] = acc;
    }
}

// ---------------------------------------------------------------------------
// Kernel 2: MLP  out = relu(X @ W1^T + b1) @ W2^T + b2   via V_WMMA_F32_16X16X4_F32
// X: [256,128] f32 feats.  grid = 16 blocks (16 frames each), block = 256 (8 waves).
// Each wave owns 2 layer-1 N-tiles (K=128 -> 32 WMMA) and 6 layer-2 N-tiles
// (K=256 -> 64 WMMA). Full f32 precision end to end.
// ---------------------------------------------------------------------------
__global__ void __launch_bounds__(256)
k_mlp_wmma(const float* __restrict__ feats,
           const float* __restrict__ w1, const float* __restrict__ b1,
           const float* __restrict__ w2, const float* __restrict__ b2,
           float* __restrict__ out) {
    __shared__ float Xs[16][128];   // input tile (8 KB)
    __shared__ float Hs[16][256];   // hidden tile (16 KB)

    const int row0 = blockIdx.x * 16;
    const int tid  = threadIdx.x;
    const int wave = tid >> 5;      // 0..7
    const int lane = tid & 31;

    // Stage X tile
    for (int idx = tid; idx < 16 * 128; idx += 256)
        Xs[idx >> 7][idx & 127] = feats[(row0 + (idx >> 7)) * 128 + (idx & 127)];
    __syncthreads();

    // WMMA f32 16x16x4 lane mapping (ISA 7.12.2):
    //  A: lane -> m = lane&15 ; VGPR0/1 hold k = 2*(lane>>4), k+1
    //  B: lane -> n = lane&15 ; VGPR0/1 hold k = 2*(lane>>4), k+1
    //  C/D: VGPR v -> row v + 8*(lane>>4), col lane&15
    const int m    = lane & 15;
    const int koff = (lane >> 4) * 2;
    const int mhi  = (lane >> 4) * 8;

    // ---- Layer 1: H = relu(X @ W1^T + b1), N = 256 -> 16 tiles, 2 per wave
    #pragma unroll
    for (int t = 0; t < 2; ++t) {
        const int n = (wave * 2 + t) * 16 + (lane & 15);
        v8f acc = {};
        for (int kb = 0; kb < 128; kb += 4) {
            v2f a, b;
            a.x = Xs[m][kb + koff];
            a.y = Xs[m][kb + koff + 1];
            b.x = w1[n * 128 + kb + koff];       // B[k][n] = W1[n][k]
            b.y = w1[n * 128 + kb + koff + 1];
            acc = __builtin_amdgcn_wmma_f32_16x16x4_f32(
                false, a, false, b, (short)0, acc, false, false);
        }
        const float bias = b1[n];
        #pragma unroll
        for (int v = 0; v < 8; ++v)
            Hs[mhi + v][n] = fmaxf(acc[v] + bias, 0.0f);  // fused bias+ReLU
    }
    __syncthreads();

    // ---- Layer 2: OUT = H @ W2^T + b2, N = 768 -> 48 tiles, 6 per wave
    #pragma unroll
    for (int t = 0; t < 6; ++t) {
        const int n = (wave * 6 + t) * 16 + (lane & 15);
        v8f acc = {};
        for (int kb = 0; kb < 256; kb += 4) {
            v2f a, b;
            a.x = Hs[m][kb + koff];
            a.y = Hs[m][kb + koff + 1];
            b.x = w2[n * 256 + kb + koff];       // B[k][n] = W2[n][k]
            b.y = w2[n * 256 + kb + koff + 1];
            acc = __builtin_amdgcn_wmma_f32_16x16x4_f32(
                false, a, false, b, (short)0, acc, false, false);
        }
        const float bias = b2[n];
        #pragma unroll
        for (int v = 0; v < 8; ++v)
            out[(size_t)(row0 + mhi + v) * 768 + n] = acc[v] + bias;
    }
}

extern "C" void kernel_launch(void* const* d_in, const int* in_sizes, int n_in,
                              void* d_out, int out_size, void* d_ws, size_t ws_size,
                              hipStream_t stream) {
    const float* images = (const float*)d_in[0];  // [8,32,3,224,224]
    const float* w1     = (const float*)d_in[1];  // [256,128]
    const float* b1     = (const float*)d_in[2];  // [256]
    const float* w2     = (const float*)d_in[3];  // [768,256]
    const float* b2     = (const float*)d_in[4];  // [768]
    float*       outp   = (float*)d_out;          // [8,32,768]
    float*       feats  = (float*)d_ws;           // 256*128 f32 scratch (128 KB)

    const int frames = 8 * 32;                    // B*T = 256
    k_gray_dct_feats<<<frames, NPIX, 0, stream>>>(images, feats);
    k_mlp_wmma<<<frames / 16, 256, 0, stream>>>(feats, w1, b1, w2, b2, outp);
}